// GraphSAGEClassifier_74251394614039
// MI455X (gfx1250) — compile-verified
//
#include <hip/hip_runtime.h>
#include <hip/hip_bf16.h>

// ---------------------------------------------------------------------------
// GraphSAGE (2x SAGEConv + mean-pool + linear) for MI455X / gfx1250.
//
// Strategy (see roofline reasoning): edge aggregation via f32 global atomics
// (dest arrays fit in the 192MB L2), GEMMs via V_WMMA_F32_16X16X4_F32 with
// activation tiles staged in LDS (bank-conflict-free [k][17] layout).
// ---------------------------------------------------------------------------

#define N_NODES   100000
#define IN_DIM    128
#define HIDDEN    256
#define N_CLASSES 16
#define N_EDGES   1600000
#define N_GRAPHS  64

typedef __attribute__((ext_vector_type(2))) float v2f;
typedef __attribute__((ext_vector_type(8))) float v8f;

// --------------------------------- utility ---------------------------------

__global__ void zero_kernel(float* __restrict__ p, long n) {
    long t = (long)blockIdx.x * blockDim.x + threadIdx.x;
    if (t < n) p[t] = 0.0f;
}

__global__ void degree_kernel(const int* __restrict__ dst,
                              float* __restrict__ deg, long nEdges) {
    long e = (long)blockIdx.x * blockDim.x + threadIdx.x;
    if (e < nEdges) atomicAdd(&deg[dst[e]], 1.0f);
}

__global__ void invdeg_kernel(const float* __restrict__ deg,
                              float* __restrict__ invdeg, int n) {
    int t = blockIdx.x * blockDim.x + threadIdx.x;
    if (t < n) invdeg[t] = 1.0f / fmaxf(deg[t], 1.0f);
}

// -------------------------- edge scatter (sum) -----------------------------
// One thread per (edge, 4-float chunk): coalesced float4 gather from src row,
// 4x global_atomic_add_f32 into dst row (resolves in L2; agg arrays fit).

__global__ void scatter_add_kernel(const float* __restrict__ feat,
                                   const int* __restrict__ src,
                                   const int* __restrict__ dst,
                                   float* __restrict__ agg,
                                   long total, int D, int cshift) {
    long t = (long)blockIdx.x * blockDim.x + threadIdx.x;
    if (t >= total) return;
    long e  = t >> cshift;
    int  ch = (int)(t & ((1 << cshift) - 1));
    int  s  = src[e];
    int  d  = dst[e];
    const float4 v = *(const float4*)(feat + (size_t)s * D + (ch << 2));
    float* ap = agg + (size_t)d * D + (ch << 2);
    atomicAdd(ap + 0, v.x);
    atomicAdd(ap + 1, v.y);
    atomicAdd(ap + 2, v.z);
    atomicAdd(ap + 3, v.w);
}

// ------------------------- fused SAGEConv (WMMA) ---------------------------
// out[n,0:256] = relu( (agg[n,:]*invdeg[n]) @ Wl + x[n,:] @ Wr + bias )
// Block = 512 threads = 16 waves; block owns 16 node-rows x all 256 cols.
// Each wave computes one 16x16 f32 tile with V_WMMA_F32_16X16X4_F32.
// A-tiles (scaled agg + root x) staged in LDS as [k][17] (conflict-free).

__global__ void __launch_bounds__(512)
sage_wmma_kernel(const float* __restrict__ agg,
                 const float* __restrict__ invdeg,
                 const float* __restrict__ xroot,
                 const float* __restrict__ Wl,    // [K, 256] row-major
                 const float* __restrict__ Wr,    // [K, 256] row-major
                 const float* __restrict__ bias,  // [256]
                 float* __restrict__ out,         // [N, 256]
                 int K, int kshift) {
    __shared__ float sA[2][HIDDEN][17];          // [mean|x][k][row(+pad)]

    const int m0   = blockIdx.x * 16;
    const int lane = threadIdx.x & 31;
    const int wave = threadIdx.x >> 5;           // 0..15 -> col tile
    const int kmask = K - 1;

    // ---- cooperative stage: 16 rows x K cols of both A matrices ----
    for (int t = threadIdx.x; t < 16 * K; t += blockDim.x) {
        int r = t >> kshift;                     // node row within tile
        int k = t & kmask;
        size_t g = (size_t)(m0 + r) * K + k;
        sA[0][k][r] = agg[g] * invdeg[m0 + r];   // mean neighbors
        sA[1][k][r] = xroot[g];                  // root features
    }
    __syncthreads();

    // ---- WMMA accumulation: D = A(16xK) x B(Kx16), K in steps of 4 ----
    const int mrow  = lane & 15;                 // M index (A), N index (B)
    const int khalf = (lane >> 4) << 1;          // lanes 16-31 hold K+2,K+3
    const int col   = (wave << 4) + (lane & 15); // output column

    v8f c = {};
    for (int h = 0; h < 2; ++h) {
        const float* __restrict__ W = (h == 0) ? Wl : Wr;
        for (int kk = 0; kk < K; kk += 4) {
            const int ka = kk + khalf;
            v2f a, b;
            a.x = sA[h][ka][mrow];
            a.y = sA[h][ka + 1][mrow];
            b.x = W[(size_t)ka * HIDDEN + col];
            b.y = W[(size_t)(ka + 1) * HIDDEN + col];
            c = __builtin_amdgcn_wmma_f32_16x16x4_f32(
                    false, a, false, b, (short)0, c, false, false);
        }
    }

    // ---- epilogue: bias + relu, C layout: vgpr r -> M=r+8*(lane/16) ----
    const float bb = bias[col];
    const int mhi = (lane >> 4) << 3;
    #pragma unroll
    for (int r = 0; r < 8; ++r) {
        int orow = m0 + r + mhi;
        float v = c[r] + bb;
        out[(size_t)orow * HIDDEN + col] = v > 0.0f ? v : 0.0f;
    }
}

// --------------------------- global mean pool ------------------------------

__global__ void pool_sum_kernel(const float* __restrict__ h,
                                const int* __restrict__ batch,
                                float* __restrict__ psum,   // [64,256]
                                float* __restrict__ pcnt) { // [64]
    long t = (long)blockIdx.x * blockDim.x + threadIdx.x;
    if (t >= (long)N_NODES * (HIDDEN / 4)) return;
    int n  = (int)(t >> 6);                      // HIDDEN/4 = 64 chunks
    int ch = (int)(t & 63);
    int g  = batch[n];
    const float4 v = *(const float4*)(h + (size_t)n * HIDDEN + (ch << 2));
    float* ap = psum + (size_t)g * HIDDEN + (ch << 2);
    atomicAdd(ap + 0, v.x);
    atomicAdd(ap + 1, v.y);
    atomicAdd(ap + 2, v.z);
    atomicAdd(ap + 3, v.w);
    if (ch == 0) atomicAdd(&pcnt[g], 1.0f);
}

__global__ void classify_kernel(const float* __restrict__ psum,
                                const float* __restrict__ pcnt,
                                const float* __restrict__ fcW,  // [256,16]
                                const float* __restrict__ fcb,  // [16]
                                float* __restrict__ outp) {
    int t = blockIdx.x * blockDim.x + threadIdx.x;
    if (t >= N_GRAPHS * N_CLASSES) return;
    int g = t >> 4, cls = t & 15;
    float inv = 1.0f / fmaxf(pcnt[g], 1.0f);
    float acc = fcb[cls];
    for (int hh = 0; hh < HIDDEN; ++hh)
        acc += psum[g * HIDDEN + hh] * inv * fcW[hh * N_CLASSES + cls];
    outp[t] = acc;
}

// --------------------------------- launch ----------------------------------

extern "C" void kernel_launch(void* const* d_in, const int* in_sizes, int n_in,
                              void* d_out, int out_size, void* d_ws, size_t ws_size,
                              hipStream_t stream) {
    const float* x    = (const float*)d_in[0];
    const int*   ei   = (const int*)d_in[1];     // [2, N_EDGES]
    const int*   bat  = (const int*)d_in[2];
    const float* W1l  = (const float*)d_in[3];
    const float* b1   = (const float*)d_in[4];
    const float* W1r  = (const float*)d_in[5];
    const float* W2l  = (const float*)d_in[6];
    const float* b2   = (const float*)d_in[7];
    const float* W2r  = (const float*)d_in[8];
    const float* fcW  = (const float*)d_in[9];
    const float* fcb  = (const float*)d_in[10];
    float* outp = (float*)d_out;

    const int* src = ei;
    const int* dst = ei + N_EDGES;

    // workspace layout (floats); zeroed region first (~154MB, total ~359MB)
    float* agg1 = (float*)d_ws;                               // N*128
    float* agg2 = agg1 + (size_t)N_NODES * IN_DIM;            // N*256
    float* deg  = agg2 + (size_t)N_NODES * HIDDEN;            // N
    float* psum = deg + N_NODES;                              // 64*256
    float* pcnt = psum + (size_t)N_GRAPHS * HIDDEN;           // 64
    float* invd = pcnt + N_GRAPHS;                            // N
    float* h1   = invd + N_NODES;                             // N*256
    float* h2   = h1 + (size_t)N_NODES * HIDDEN;              // N*256

    const long nZero = (long)N_NODES * IN_DIM + (long)N_NODES * HIDDEN +
                       N_NODES + (long)N_GRAPHS * HIDDEN + N_GRAPHS;
    zero_kernel<<<(unsigned)((nZero + 255) / 256), 256, 0, stream>>>(agg1, nZero);

    degree_kernel<<<(N_EDGES + 255) / 256, 256, 0, stream>>>(dst, deg, N_EDGES);
    invdeg_kernel<<<(N_NODES + 255) / 256, 256, 0, stream>>>(deg, invd, N_NODES);

    // ---- layer 1 ----
    {
        const long total = (long)N_EDGES * (IN_DIM / 4);      // cshift=5
        scatter_add_kernel<<<(unsigned)((total + 255) / 256), 256, 0, stream>>>(
            x, src, dst, agg1, total, IN_DIM, 5);
        sage_wmma_kernel<<<N_NODES / 16, 512, 0, stream>>>(
            agg1, invd, x, W1l, W1r, b1, h1, IN_DIM, 7);
    }

    // ---- layer 2 ----
    {
        const long total = (long)N_EDGES * (HIDDEN / 4);      // cshift=6
        scatter_add_kernel<<<(unsigned)((total + 255) / 256), 256, 0, stream>>>(
            h1, src, dst, agg2, total, HIDDEN, 6);
        sage_wmma_kernel<<<N_NODES / 16, 512, 0, stream>>>(
            agg2, invd, h1, W2l, W2r, b2, h2, HIDDEN, 8);
    }

    // ---- pool + classify ----
    {
        const long total = (long)N_NODES * (HIDDEN / 4);
        pool_sum_kernel<<<(unsigned)((total + 255) / 256), 256, 0, stream>>>(
            h2, bat, psum, pcnt);
        classify_kernel<<<(N_GRAPHS * N_CLASSES + 255) / 256, 256, 0, stream>>>(
            psum, pcnt, fcW, fcb, outp);
    }
}